// OffSmoothL1LossPlus_54417235640819
// MI455X (gfx1250) — compile-verified
//
#include <hip/hip_runtime.h>

// Problem constants (from reference)
#define BB    32
#define CC    10
#define HH    256
#define WW    256
#define KK    500
#define NCLS  80
#define HWSZ  (HH * WW)          // 65536
#define NPTS  (BB * KK)          // 16000
#define BLOCK 256
#define NBLK  ((NPTS + BLOCK - 1) / BLOCK)   // 63

typedef float v2f __attribute__((ext_vector_type(2)));
typedef float v8f __attribute__((ext_vector_type(8)));

// Exact f32 wave32 sum using the matrix pipe.
// A (16x4 f32): lane L<16 -> A[L][0]=v, A[L][1]=0 ; lane L>=16 -> A[L-16][2]=v, A[L-16][3]=0.
// B = ones(4x16)  =>  D[m][n] = v_m + v_{m+16} (row value replicated across columns).
// D layout: lanes 0-15, VGPR r = q_r ; lanes 16-31, VGPR r = q_{8+r}.
// Sum the 8 D VGPRs per lane, then fold the two 16-lane halves with one shfl_xor.
__device__ __forceinline__ float wave_sum32(float v) {
    v2f a; a[0] = v;    a[1] = 0.0f;
    v2f b; b[0] = 1.0f; b[1] = 1.0f;
    v8f c = {};
    v8f d = __builtin_amdgcn_wmma_f32_16x16x4_f32(
        /*neg_a=*/false, a, /*neg_b=*/false, b,
        /*c_mod=*/(short)0, c, /*reuse_a=*/false, /*reuse_b=*/false);
    float s = ((d[0] + d[1]) + (d[2] + d[3])) + ((d[4] + d[5]) + (d[6] + d[7]));
    s += __shfl_xor(s, 16, 32);
    return s;   // total in every lane
}

__device__ __forceinline__ float smooth_l1(float diff) {
    float ad = fabsf(diff);
    return (ad < 1.0f) ? (0.5f * diff * diff) : (ad - 0.5f);
}

__global__ void __launch_bounds__(BLOCK)
off_smooth_l1_partials(const float* __restrict__ output,       // [B,C,H,W]
                       const int* __restrict__ mask,           // [B,K]
                       const long long* __restrict__ ind,      // [B,K]
                       const float* __restrict__ target,       // [B,K,C]
                       const long long* __restrict__ inde,     // [B,K,3]
                       const float* __restrict__ hm,           // [B,NCLS,H,W]
                       float2* __restrict__ partials) {        // [NBLK]
    const int idx = blockIdx.x * BLOCK + threadIdx.x;

    float lossv = 0.0f, cntv = 0.0f;
    if (idx < NPTS) {
        const int b   = idx / KK;
        const int pos = (int)ind[idx];               // [0, H*W)
        const float m = (float)mask[idx];

        const float* outb = output + (size_t)b * CC * HWSZ + pos;
        const float* tg   = target + (size_t)idx * CC;

        // Issue all 10 gathers independently; compiler pipelines the loads.
        float pred[CC];
        float tv[CC];
#pragma unroll
        for (int ch = 0; ch < CC; ++ch) pred[ch] = outb[(size_t)ch * HWSZ];
#pragma unroll
        for (int ch = 0; ch < CC; ++ch) tv[ch] = tg[ch];

        float acc = 0.0f;
#pragma unroll
        for (int ch = 0; ch < CC; ++ch) acc += smooth_l1(pred[ch] - tv[ch]);
        const float per = acc * (1.0f / (float)CC);

        const long long* ie = inde + (size_t)idx * 3;
        const int i0 = (int)ie[0];
        const int i1 = (int)ie[1];
        const int i2 = (int)ie[2];
        const float p = hm[((size_t)b * NCLS + i0) * HWSZ + (size_t)i1 * WW + i2];
        float w = 1.0f + p;
        w = w * w;

        lossv = per * w * m;
        cntv  = m;
    }

    // Wave-level exact f32 reduction on the matrix pipe (EXEC is all-1s here).
    const float ls = wave_sum32(lossv);
    const float cs = wave_sum32(cntv);

    __shared__ float sL[BLOCK / 32];
    __shared__ float sC[BLOCK / 32];
    const int wave = threadIdx.x >> 5;
    const int lane = threadIdx.x & 31;
    if (lane == 0) { sL[wave] = ls; sC[wave] = cs; }
    __syncthreads();

    if (threadIdx.x == 0) {
        float l = 0.0f, c = 0.0f;
#pragma unroll
        for (int i = 0; i < BLOCK / 32; ++i) { l += sL[i]; c += sC[i]; }
        partials[blockIdx.x] = make_float2(l, c);   // deterministic: private slot, no atomics
    }
}

__global__ void __launch_bounds__(32)
off_smooth_l1_finalize(const float2* __restrict__ partials, float* __restrict__ out) {
    float l = 0.0f, c = 0.0f;
    for (int j = threadIdx.x; j < NBLK; j += 32) {
        const float2 v = partials[j];
        l += v.x;
        c += v.y;
    }
    // Reconverged: full wave active at the WMMA.
    const float ls = wave_sum32(l);
    const float cs = wave_sum32(c);
    if (threadIdx.x == 0) {
        const float loss = ls / fmaxf(cs, 1.0f);
        out[0] = (cs > 0.0f) ? loss : 0.0f;
    }
}

extern "C" void kernel_launch(void* const* d_in, const int* in_sizes, int n_in,
                              void* d_out, int out_size, void* d_ws, size_t ws_size,
                              hipStream_t stream) {
    (void)in_sizes; (void)n_in; (void)out_size; (void)ws_size;
    const float*     output = (const float*)d_in[0];
    const int*       mask   = (const int*)d_in[1];
    const long long* ind    = (const long long*)d_in[2];
    const float*     target = (const float*)d_in[3];
    const long long* inde   = (const long long*)d_in[4];
    const float*     hm     = (const float*)d_in[5];

    float2* partials = (float2*)d_ws;   // NBLK * 8 bytes

    off_smooth_l1_partials<<<NBLK, BLOCK, 0, stream>>>(output, mask, ind, target, inde, hm, partials);
    off_smooth_l1_finalize<<<1, 32, 0, stream>>>(partials, (float*)d_out);
}